// QuantumRecurrentUnit_86741159510238
// MI455X (gfx1250) — compile-verified
//
#include <hip/hip_runtime.h>
#include <cstdint>

// Closed form of the reference quantum circuit (see analysis):
//   out[row, w] = prod_{j<=w} cos(weights[j]) * cos(x[row, j])
// x rows are 16 floats, only the first 8 are used; out rows are 8 floats.
//
// CDNA5 path: per-lane async global->LDS staging (global_load_async_to_lds_b128,
// tracked by ASYNCcnt) + s_wait_asynccnt. Each lane reads back only its own
// staged bytes, so no workgroup barrier is required.

#define NQ 8
static constexpr float kInv2Pi = 0.15915494309189535f;  // v_cos_f32 takes turns

__global__ __launch_bounds__(256) void qru_closed_form(
    const float* __restrict__ x,        // (nrows, 16)
    const float* __restrict__ weights,  // (8,)
    float* __restrict__ out,            // (nrows, 8)
    int nrows) {
  __shared__ __align__(16) float tile[256 * NQ];  // 8 KiB: 32 B per thread/row

  const int tid = threadIdx.x;
  const int row = blockIdx.x * 256 + tid;
  if (row >= nrows) return;

  // ---- Async stage: first 32 B (the 8 angle floats) of this row into LDS.
  {
    const unsigned long long gaddr =
        (unsigned long long)(uintptr_t)(x + (size_t)row * 16u);
    // Flat shared address low 32 bits == LDS byte offset (aperture spec).
    const unsigned lds_off =
        (unsigned)(uintptr_t)(&tile[0]) + (unsigned)tid * 32u;
    asm volatile(
        "global_load_async_to_lds_b128 %0, %1, off\n\t"
        "global_load_async_to_lds_b128 %0, %1, off offset:16"
        :: "v"(lds_off), "v"(gaddr)
        : "memory");
  }

  // ---- Per-wire constant: W[w] = prod_{j<=w} cos(weights[j]).
  // Issued before the async wait so the scalar/global loads overlap it.
  float cw[NQ];
  {
    float p = 1.0f;
#pragma unroll
    for (int j = 0; j < NQ; ++j) {
      p *= __builtin_amdgcn_cosf(weights[j] * kInv2Pi);
      cw[j] = p;
    }
  }

  // Wait for this wave's async copies to land in LDS.
  asm volatile("s_wait_asynccnt 0" ::: "memory");

  const float4* tp = (const float4*)(&tile[tid * NQ]);
  const float4 a0 = tp[0];
  const float4 a1 = tp[1];
  const float a[NQ] = {a0.x, a0.y, a0.z, a0.w, a1.x, a1.y, a1.z, a1.w};

  // Cumulative product of cos(angle_j) * cumulative weight product.
  float o[NQ];
  float p = 1.0f;
#pragma unroll
  for (int j = 0; j < NQ; ++j) {
    p *= __builtin_amdgcn_cosf(a[j] * kInv2Pi);
    o[j] = p * cw[j];
  }

  float4* op = (float4*)(out + (size_t)row * NQ);
  op[0] = make_float4(o[0], o[1], o[2], o[3]);
  op[1] = make_float4(o[4], o[5], o[6], o[7]);
}

extern "C" void kernel_launch(void* const* d_in, const int* in_sizes, int n_in,
                              void* d_out, int out_size, void* d_ws,
                              size_t ws_size, hipStream_t stream) {
  const float* x = (const float*)d_in[0];        // (256,128,16) f32
  const float* w = (const float*)d_in[1];        // (8,) f32
  float* out = (float*)d_out;                    // (256,128,8) f32
  const int nrows = in_sizes[0] / 16;            // 32768
  const int blocks = (nrows + 255) / 256;        // 128 blocks of 8 waves
  qru_closed_form<<<blocks, 256, 0, stream>>>(x, w, out, nrows);
}